// TokensChooseMaskedRouter_81106162417786
// MI455X (gfx1250) — compile-verified
//
#include <hip/hip_runtime.h>
#include <hip/hip_bf16.h>

typedef __attribute__((ext_vector_type(2))) float v2f;
typedef __attribute__((ext_vector_type(8))) float v8f;

#define GG 4
#define TT 4096
#define DD 2048
#define EE 64
#define CC 128
#define NTOK (GG*TT)              // 16384
#define DISP_ELEMS (134217728L)   // 4*4096*64*128

// ---------------------------------------------------------------------------
// Kernel A: logits = X[16384,2048] * W[2048,64] + b  via v_wmma_f32_16x16x4_f32
// Block = 128 threads (4 waves). Block computes 64 rows x 64 cols.
// X tile stored [m][k] (stride 68), W tile stored TRANSPOSED [n][k] (stride 68)
// so both A and B fragments are contiguous aligned 8-byte LDS reads
// (ds_load_2addr_b64), no re-pack moves, fewer dscnt waits.
// ---------------------------------------------------------------------------
__global__ __launch_bounds__(128) void gemm_logits_kernel(
    const float* __restrict__ X, const float* __restrict__ W,
    const float* __restrict__ bias, float* __restrict__ logits)
{
  __shared__ float xs[64][68];   // [m][k]
  __shared__ float wt[64][68];   // [n][k]  (transposed W tile)
  const int tid  = threadIdx.x;
  const int wave = tid >> 5;
  const int lane = tid & 31;
  const int l15  = lane & 15;
  const long rowBase = (long)blockIdx.x * 64;

  v8f zero = {};
  v8f acc[4];
#pragma unroll
  for (int nt = 0; nt < 4; ++nt) acc[nt] = zero;

  const int mrow = (wave << 4) + l15;           // LDS row for A fragment
  const int koff = (lane >> 4) << 1;            // lanes 16-31 hold K+2,K+3

  for (int k0 = 0; k0 < DD; k0 += 64) {
    // cooperative tile loads (coalesced global reads along the fast dim)
#pragma unroll
    for (int it = 0; it < 32; ++it) {
      int idx = tid + it * 128;
      int r = idx >> 6, c = idx & 63;
      xs[r][c] = X[(rowBase + r) * DD + k0 + c];  // r = m, c = k
      wt[c][r] = W[(long)(k0 + r) * EE + c];      // r = k, c = n (transpose)
    }
    // start pulling the next X tile toward L2/L0 while we compute this one
    if (k0 + 64 < DD) {
      int r = tid >> 1, c = (tid & 1) << 5;       // 2 prefetches per row strip
      __builtin_prefetch(&X[(rowBase + r) * DD + k0 + 64 + c], 0, 3);
    }
    __syncthreads();

#pragma unroll
    for (int kk = 0; kk < 64; kk += 4) {
      int kb = kk + koff;                         // even -> 8B aligned
      v2f a = *reinterpret_cast<const v2f*>(&xs[mrow][kb]);
#pragma unroll
      for (int nt = 0; nt < 4; ++nt) {
        int n = (nt << 4) + l15;
        v2f b = *reinterpret_cast<const v2f*>(&wt[n][kb]);
        acc[nt] = __builtin_amdgcn_wmma_f32_16x16x4_f32(
            false, a, false, b, (short)0, acc[nt], false, false);
      }
    }
    __syncthreads();
  }

  // C/D layout: VGPR v -> M=v (lanes 0-15), M=v+8 (lanes 16-31); N = lane&15
  const int mhalf = (lane >> 4) << 3;
#pragma unroll
  for (int nt = 0; nt < 4; ++nt) {
    int col = (nt << 4) + l15;
    float bv = bias[col];
#pragma unroll
    for (int v = 0; v < 8; ++v) {
      long row = rowBase + (wave << 4) + mhalf + v;
      logits[row * EE + col] = acc[nt][v] + bv;
    }
  }
}

// ---------------------------------------------------------------------------
// Kernel B: softmax over E=64, top-2 (jax tie semantics: lowest index wins),
// accumulate z-loss and per-(g,e) prob-sums / expert-hit counts.
// One wave32 per token (2 experts per lane), 8 tokens per 256-thread block.
// ---------------------------------------------------------------------------
__global__ __launch_bounds__(256) void softmax_topk_kernel(
    const float* __restrict__ logits, float* __restrict__ gates,
    int* __restrict__ experts, float* __restrict__ sum_probs,
    float* __restrict__ cnt_em, float* __restrict__ zsum)
{
  __shared__ float sp[64];
  __shared__ float sc[64];
  const int tid = threadIdx.x, wave = tid >> 5, lane = tid & 31;
  if (tid < 64) { sp[tid] = 0.f; sc[tid] = 0.f; }
  __syncthreads();

  const long tok = (long)blockIdx.x * 8 + wave;
  const float* L = logits + tok * EE;
  float l0 = L[lane], l1 = L[lane + 32];

  float m = fmaxf(l0, l1);
  for (int off = 16; off; off >>= 1) m = fmaxf(m, __shfl_xor(m, off, 32));
  float e0 = __expf(l0 - m), e1 = __expf(l1 - m);
  float s = e0 + e1;
  for (int off = 16; off; off >>= 1) s += __shfl_xor(s, off, 32);
  float inv = 1.0f / s;
  float p0 = e0 * inv, p1 = e1 * inv;

  float logS = __logf(s);
  float ls0 = (l0 - m) - logS, ls1 = (l1 - m) - logS;
  float z = ls0 * ls0 + ls1 * ls1;
  for (int off = 16; off; off >>= 1) z += __shfl_xor(z, off, 32);

  atomicAdd(&sp[lane], p0);
  atomicAdd(&sp[lane + 32], p1);

  // local order (lower expert index first on ties)
  float av, bv; int ai, bi;
  if (p0 >= p1) { av = p0; ai = lane;      bv = p1; bi = lane + 32; }
  else          { av = p1; ai = lane + 32; bv = p0; bi = lane;      }

  float mv = av; int mi = ai;
  for (int off = 16; off; off >>= 1) {
    float ov = __shfl_xor(mv, off, 32);
    int   oi = __shfl_xor(mi, off, 32);
    if (ov > mv || (ov == mv && oi < mi)) { mv = ov; mi = oi; }
  }
  float cv = (ai == mi) ? bv : av;
  int   ci = (ai == mi) ? bi : ai;
  float m2v = cv; int m2i = ci;
  for (int off = 16; off; off >>= 1) {
    float ov = __shfl_xor(m2v, off, 32);
    int   oi = __shfl_xor(m2i, off, 32);
    if (ov > m2v || (ov == m2v && oi < m2i)) { m2v = ov; m2i = oi; }
  }

  if (lane == 0) {
    gates[tok * 2]     = mv;
    gates[tok * 2 + 1] = m2v;
    experts[tok * 2]     = mi;
    experts[tok * 2 + 1] = m2i;
    atomicAdd(&sc[mi], 1.0f);
    atomicAdd(&sc[m2i], 1.0f);
    atomicAdd(zsum, z);
  }
  __syncthreads();
  const int gblk = (int)(((long)blockIdx.x * 8) >> 12);
  if (tid < 64) {
    atomicAdd(&sum_probs[gblk * 64 + tid], sp[tid]);
    atomicAdd(&cnt_em[gblk * 64 + tid], sc[tid]);
  }
}

// ---------------------------------------------------------------------------
// Kernel C: per-group stable descending sort by top-1 gate (BPR).
// 64-bit key = (order-preserving float bits << 32) | ~index  -> matches
// jnp.argsort(-gate) tie behavior. In-LDS bitonic, 4096 keys, 1024 threads.
// ---------------------------------------------------------------------------
__global__ __launch_bounds__(1024) void bpr_sort_kernel(
    const float* __restrict__ gates, int* __restrict__ perm)
{
  __shared__ unsigned long long key[4096];
  const int g = blockIdx.x, tid = threadIdx.x;
  for (int i = tid; i < 4096; i += 1024) {
    float gv = gates[((long)g * TT + i) * 2];
    unsigned u = __float_as_uint(gv);
    unsigned ord = (u & 0x80000000u) ? ~u : (u | 0x80000000u);
    key[i] = ((unsigned long long)ord << 32) | (unsigned)(~(unsigned)i);
  }
  __syncthreads();
  for (int k = 2; k <= 4096; k <<= 1) {
    for (int j = k >> 1; j > 0; j >>= 1) {
      for (int i = tid; i < 4096; i += 1024) {
        int ixj = i ^ j;
        if (ixj > i) {
          unsigned long long a = key[i], b = key[ixj];
          bool swp = ((i & k) != 0) ? (a > b) : (a < b);  // descending overall
          if (swp) { key[i] = b; key[ixj] = a; }
        }
      }
      __syncthreads();
    }
  }
  for (int i = tid; i < 4096; i += 1024)
    perm[g * TT + i] = (int)(~(unsigned)(key[i] & 0xFFFFFFFFull));
}

// ---------------------------------------------------------------------------
// Kernel D: capacity position = rank of each (k,sorted-token) entry within its
// expert's ordered list (all k=0 entries before k=1). Block per (g,e):
// count per thread-chunk, LDS inclusive scan, then rank assignment.
// ---------------------------------------------------------------------------
__global__ __launch_bounds__(256) void priority_kernel(
    const int* __restrict__ perm, const int* __restrict__ experts,
    int* __restrict__ prio)
{
  __shared__ int cnts[256];
  const int g = blockIdx.x >> 6;
  const int e = blockIdx.x & 63;
  const int tid = threadIdx.x;
  const int base = tid * 32;              // 8192 entries / 256 threads

  int cnt = 0;
  for (int jj = 0; jj < 32; ++jj) {
    int j = base + jj;
    int i = j & 4095, k = j >> 12;
    int token = perm[g * TT + i];
    int ej = experts[((long)g * TT + token) * 2 + k];
    cnt += (ej == e);
  }
  cnts[tid] = cnt;
  __syncthreads();
  for (int off = 1; off < 256; off <<= 1) {
    int t = (tid >= off) ? cnts[tid - off] : 0;
    __syncthreads();
    cnts[tid] += t;
    __syncthreads();
  }
  int run = cnts[tid] - cnt;              // exclusive prefix
  for (int jj = 0; jj < 32; ++jj) {
    int j = base + jj;
    int i = j & 4095, k = j >> 12;
    int token = perm[g * TT + i];
    long ti = (long)g * TT + token;
    int ej = experts[ti * 2 + k];
    if (ej == e) prio[ti * 2 + k] = run++;
  }
}

// ---------------------------------------------------------------------------
// Output kernels: vectorized zero-fill of the 1 GB dense output, then scatter
// the <=2 nonzeros per token, then scalar losses.
// ---------------------------------------------------------------------------
__global__ void fill_zero_kernel(float4* __restrict__ out, long n4)
{
  long i = (long)blockIdx.x * blockDim.x + threadIdx.x;
  long stride = (long)gridDim.x * blockDim.x;
  float4 z = make_float4(0.f, 0.f, 0.f, 0.f);
  for (; i < n4; i += stride) out[i] = z;
}

__global__ __launch_bounds__(256) void scatter_kernel(
    const int* __restrict__ prio, const int* __restrict__ experts,
    const float* __restrict__ gates, float* __restrict__ out)
{
  long j = (long)blockIdx.x * 256 + threadIdx.x;   // 0..32767 = (tok,k)
  long tok = j >> 1;
  int p = prio[j];
  if (p >= 0 && p < CC) {
    int e = experts[j];
    float gv = gates[j];
    long off = (tok * EE + e) * CC + p;
    out[off] = 1.0f;                 // dispatch
    out[DISP_ELEMS + off] = gv;      // combine = probs at chosen (e,c)
  }
}

__global__ __launch_bounds__(256) void finalize_kernel(
    const float* __restrict__ sum_probs, const float* __restrict__ cnt_em,
    const float* __restrict__ zsum, float* __restrict__ out)
{
  __shared__ float red[256];
  int tid = threadIdx.x;
  float term = (cnt_em[tid] * (1.0f / TT)) * (sum_probs[tid] * (1.0f / TT));
  red[tid] = term;
  __syncthreads();
  for (int off = 128; off; off >>= 1) {
    if (tid < off) red[tid] += red[tid + off];
    __syncthreads();
  }
  if (tid == 0) {
    // aux = sum_{g,e} term * E^2 / (G*E) = sum * 4096/256
    out[2 * DISP_ELEMS]     = red[0] * 16.0f;
    out[2 * DISP_ELEMS + 1] = zsum[0] * (1.0f / ((float)GG * TT * EE));
  }
}

__global__ void init_ws_kernel(float* __restrict__ sum_probs,
                               float* __restrict__ cnt_em,
                               float* __restrict__ zsum)
{
  int t = threadIdx.x;
  if (t < 256) { sum_probs[t] = 0.f; cnt_em[t] = 0.f; }
  if (t == 0) zsum[0] = 0.f;
}

// ---------------------------------------------------------------------------
extern "C" void kernel_launch(void* const* d_in, const int* in_sizes, int n_in,
                              void* d_out, int out_size, void* d_ws, size_t ws_size,
                              hipStream_t stream) {
  const float* X    = (const float*)d_in[0];   // [4,4096,2048]
  const float* W    = (const float*)d_in[1];   // [2048,64]
  const float* bias = (const float*)d_in[2];   // [64]
  float* out = (float*)d_out;

  // workspace layout (floats)
  float* wsf       = (float*)d_ws;
  float* logits    = wsf;                         // 16384*64 = 1048576
  float* gates     = wsf + 1048576;               // 16384*2  = 32768
  int*   experts   = (int*)(gates + 32768);       // 32768 ints
  int*   perm      = experts + 32768;             // 16384 ints
  int*   prio      = perm + 16384;                // 32768 ints
  float* sum_probs = (float*)(prio + 32768);      // 256
  float* cnt_em    = sum_probs + 256;             // 256
  float* zsum      = cnt_em + 256;                // 1

  init_ws_kernel<<<1, 256, 0, stream>>>(sum_probs, cnt_em, zsum);
  fill_zero_kernel<<<8192, 256, 0, stream>>>((float4*)out, 2 * DISP_ELEMS / 4);

  gemm_logits_kernel<<<NTOK / 64, 128, 0, stream>>>(X, W, bias, logits);
  softmax_topk_kernel<<<NTOK / 8, 256, 0, stream>>>(logits, gates, experts,
                                                    sum_probs, cnt_em, zsum);
  bpr_sort_kernel<<<GG, 1024, 0, stream>>>(gates, perm);
  priority_kernel<<<GG * EE, 256, 0, stream>>>(perm, experts, prio);
  scatter_kernel<<<(NTOK * 2) / 256, 256, 0, stream>>>(prio, experts, gates, out);
  finalize_kernel<<<1, 256, 0, stream>>>(sum_probs, cnt_em, zsum, out);
}